// STANet_with_DFT_18571438588046
// MI455X (gfx1250) — compile-verified
//
#include <hip/hip_runtime.h>
#include <math.h>

// ---------------------------------------------------------------------------
// STANet_with_DFT forward for MI455X (gfx1250, wave32, WMMA).
// Live dataflow only (temporal/spatial attention products are dead in the ref).
// All GEMMs use V_WMMA_F32_16X16X4_F32 (fp32 in/out, full precision).
// Wave tile = 16x64 (4 accumulators); K templated + fully unrolled modulo-4
// fragment buffers: depth-2 prefetch with zero rotation copies and immediate
// -offset loads (no per-step address VALU, no drain-to-zero waits).
// Shapes: B=8 N=512 FIN=16 T=24 K=3 E=H=MLPH=64.
// ---------------------------------------------------------------------------

typedef float v2f __attribute__((ext_vector_type(2)));
typedef float v8f __attribute__((ext_vector_type(8)));

#define DEVINL __device__ __forceinline__

DEVINL v8f wmma4(v2f a, v2f b, v8f c) {
  // D = A(16x4) * B(4x16) + C(16x16), fp32.
  return __builtin_amdgcn_wmma_f32_16x16x4_f32(false, a, false, b, (short)0, c,
                                               false, false);
}

// Fully-unrolled K loop with modulo-4 fragment buffers and depth-2 prefetch.
// loadA(k)->v2f fragment, loadB(k,b[4]) fills the 4 column-strip fragments.
// K%4==0, K>=8.
template <int K, typename LA, typename LB>
DEVINL void gemm_pipeline(LA loadA, LB loadB, v8f acc[4]) {
  constexpr int S = K / 4;
  v2f a[4];
  v2f b[4][4];
  a[0] = loadA(0);
  loadB(0, b[0]);
  a[1] = loadA(4);
  loadB(4, b[1]);
#pragma unroll
  for (int i = 0; i < S; ++i) {
    if (i + 2 < S) {  // compile-time per unrolled instance
      a[(i + 2) & 3] = loadA(4 * (i + 2));
      loadB(4 * (i + 2), b[(i + 2) & 3]);
    }
#pragma unroll
    for (int et = 0; et < 4; ++et)
      acc[et] = wmma4(a[i & 3], b[i & 3][et], acc[et]);
  }
}

// ======================= 1. DFT layer (delay-embed + MLP) ===================
__global__ __launch_bounds__(128) void dft_kernel(
    const float* __restrict__ x, const float* __restrict__ w1,
    const float* __restrict__ b1, const float* __restrict__ w2,
    const float* __restrict__ b2, float* __restrict__ xpost,
    float* __restrict__ xr) {
  __shared__ float sw1[64 * 64];
  __shared__ float sw2[64 * 16];
  __shared__ float sb1[64];
  __shared__ float sb2[16];
  const int tid = threadIdx.x;
  for (int i = tid; i < 4096; i += 128) sw1[i] = w1[i];
  for (int i = tid; i < 1024; i += 128) sw2[i] = w2[i];
  for (int i = tid; i < 64; i += 128) sb1[i] = b1[i];
  if (tid < 16) sb2[tid] = b2[tid];
  __syncthreads();

  const int idx = blockIdx.x * 128 + tid;  // < 98304 = B*N*T
  const int bn = idx / 24, tau = idx % 24;
  const float* xrow = x + (size_t)bn * 384;  // (F=16,T=24) block

  float xd[64];
#pragma unroll
  for (int j = 0; j < 64; ++j) {
    int flat = tau * 64 + j;  // index into (F, 4*T) block
    int f = flat / 96;
    int rem = flat % 96;
    int s = rem / 24;   // delay
    int tt = rem % 24;  // shifted time
    xd[j] = (tt >= s) ? xrow[f * 24 + (tt - s)] : 0.f;
  }
  float h1[64];
#pragma unroll 4
  for (int o = 0; o < 64; ++o) {
    float a = sb1[o];
    for (int j = 0; j < 64; ++j) a += xd[j] * sw1[j * 64 + o];
    h1[o] = fmaxf(a, 0.f);
  }
#pragma unroll
  for (int ff = 0; ff < 16; ++ff) {
    float a = sb2[ff];
    for (int o = 0; o < 64; ++o) a += h1[o] * sw2[o * 16 + ff];
    xpost[(size_t)bn * 384 + ff * 24 + tau] = a;  // (B,N,F,T)
    xr[(size_t)idx * 16 + ff] = a;                // (B,N,T,F)
  }
}

// ================== generic wave-level WMMA GEMM, 16x64 tile ================
// C[M,N] = act(A[M,K] @ B + bias); A row-major lda; B row-major (K,N) ldb or
// transposed (N,K) ldb. Grid: (N/64, M/16); 32 threads (1 wave) per block.
// M%16==0, N%64==0.
template <int K, bool TRANSB, bool ACT>
__global__ __launch_bounds__(32) void gemm16x64_kernel(
    const float* __restrict__ A, const float* __restrict__ B,
    const float* __restrict__ bias, float* __restrict__ C, int lda, int ldb,
    int ldc) {
  const int lane = threadIdx.x;
  const int n0 = blockIdx.x * 64, m0 = blockIdx.y * 16;
  const int mr = lane & 15;         // A row within tile
  const int kh = (lane >> 4) << 1;  // K sub-offset per lane-half
  const int nc = lane & 15;         // B/C column within 16-wide strip
  v8f acc[4];
#pragma unroll
  for (int et = 0; et < 4; ++et) {
    float bv = bias ? bias[n0 + et * 16 + nc] : 0.f;
#pragma unroll
    for (int i = 0; i < 8; ++i) acc[et][i] = bv;
  }

  const float* Arow = A + (size_t)(m0 + mr) * lda + kh;
  auto loadA = [&](int k) {
    v2f a;
    a.x = Arow[k];
    a.y = Arow[k + 1];
    return a;
  };
  if (!TRANSB) {
    const float* Bcol = B + n0 + nc;
    auto loadB = [&](int k, v2f b[4]) {
      const int ka = k + kh;
#pragma unroll
      for (int et = 0; et < 4; ++et) {
        b[et].x = Bcol[(size_t)ka * ldb + et * 16];
        b[et].y = Bcol[(size_t)(ka + 1) * ldb + et * 16];
      }
    };
    gemm_pipeline<K>(loadA, loadB, acc);
  } else {
    const float* Bc0 = B + (size_t)(n0 + nc) * ldb + kh;
    auto loadB = [&](int k, v2f b[4]) {
#pragma unroll
      for (int et = 0; et < 4; ++et) {
        b[et].x = Bc0[(size_t)et * 16 * ldb + k];
        b[et].y = Bc0[(size_t)et * 16 * ldb + k + 1];
      }
    };
    gemm_pipeline<K>(loadA, loadB, acc);
  }

  const int rbase = m0 + ((lane >> 4) << 3);
#pragma unroll
  for (int et = 0; et < 4; ++et)
#pragma unroll
    for (int r = 0; r < 8; ++r) {
      float v = acc[et][r];
      if (ACT) v = fmaxf(v, 0.f);
      C[(size_t)(rbase + r) * ldc + n0 + et * 16 + nc] = v;
    }
}

// ================= 2a. Cheb stage A: y[b,k] = cheb[k]^T @ x[b] ==============
// C (512 x 384) = chebT(512x512) @ xpost[b](512x384); z = b*3+k
__global__ __launch_bounds__(32) void cheb_a_kernel(
    const float* __restrict__ cheb, const float* __restrict__ xpost,
    float* __restrict__ ybuf) {
  const int lane = threadIdx.x;
  const int c0 = blockIdx.x * 64;  // col (f*24+t) strip
  const int n0 = blockIdx.y * 16;  // out row (n) tile
  const int z = blockIdx.z;        // b*3+k
  const int b = z / 3, k = z % 3;
  const int mr = lane & 15, kh = (lane >> 4) << 1, nc = lane & 15;
  const float* Ck = cheb + (size_t)k * 512 * 512 + (size_t)kh * 512 + n0 + mr;
  const float* Xb = xpost + (size_t)b * 512 * 384 + (size_t)kh * 384 + c0 + nc;
  v8f acc[4];
#pragma unroll
  for (int et = 0; et < 4; ++et)
#pragma unroll
    for (int i = 0; i < 8; ++i) acc[et][i] = 0.f;

  auto loadA = [&](int kq) {
    v2f a;
    a.x = Ck[(size_t)kq * 512];
    a.y = Ck[(size_t)(kq + 1) * 512];
    return a;
  };
  auto loadB = [&](int kq, v2f bb[4]) {
#pragma unroll
    for (int et = 0; et < 4; ++et) {
      bb[et].x = Xb[(size_t)kq * 384 + et * 16];
      bb[et].y = Xb[(size_t)(kq + 1) * 384 + et * 16];
    }
  };
  gemm_pipeline<512>(loadA, loadB, acc);

  float* Y = ybuf + (size_t)z * 512 * 384;
  const int rbase = n0 + ((lane >> 4) << 3);
#pragma unroll
  for (int et = 0; et < 4; ++et)
#pragma unroll
    for (int r = 0; r < 8; ++r)
      Y[(size_t)(rbase + r) * 384 + c0 + et * 16 + nc] = acc[et][r];
}

// ====== 2b. Cheb stage B: gmat[(b*T+t)*N+n][e] = relu(sum_{k,f} y*Theta) ====
// One wave computes a full 16x64 output tile (K'=48, ka = k*16+f).
__global__ __launch_bounds__(32) void cheb_b_kernel(
    const float* __restrict__ ybuf, const float* __restrict__ Theta,
    float* __restrict__ gmat) {
  const int lane = threadIdx.x;
  const int m0 = blockIdx.x * 16;  // row = (b*24+t)*512 + n
  const int bt = m0 >> 9, n0 = m0 & 511;
  const int b = bt / 24, t = bt % 24;
  const int mr = lane & 15, kh = (lane >> 4) << 1, nc = lane & 15;
  const float* Yr = ybuf + ((size_t)b * 3 * 512 + n0 + mr) * 384 + t;
  v8f acc[4];
#pragma unroll
  for (int et = 0; et < 4; ++et)
#pragma unroll
    for (int i = 0; i < 8; ++i) acc[et][i] = 0.f;

  auto loadA = [&](int kq) {
    const int ka = kq + kh;  // ka = k*16 + f
    v2f a;
    a.x = Yr[(size_t)(ka >> 4) * 512 * 384 + (ka & 15) * 24];
    a.y = Yr[(size_t)((ka + 1) >> 4) * 512 * 384 + ((ka + 1) & 15) * 24];
    return a;
  };
  auto loadB = [&](int kq, v2f bb[4]) {
    const int ka = kq + kh;
#pragma unroll
    for (int et = 0; et < 4; ++et) {
      bb[et].x = Theta[(size_t)ka * 64 + et * 16 + nc];
      bb[et].y = Theta[(size_t)(ka + 1) * 64 + et * 16 + nc];
    }
  };
  gemm_pipeline<48>(loadA, loadB, acc);

  const int rbase = n0 + ((lane >> 4) << 3);
#pragma unroll
  for (int et = 0; et < 4; ++et)
#pragma unroll
    for (int r = 0; r < 8; ++r)
      gmat[((size_t)bt * 512 + rbase + r) * 64 + et * 16 + nc] =
          fmaxf(acc[et][r], 0.f);
}

// ============== 3. GAT attention: flash-style, one wave / 16 queries ========
// qkv layout: [p=(b*T+t)][n][192] with q=0..63, k=64..127, v=128..191
__global__ __launch_bounds__(32) void attn_kernel(const float* __restrict__ qkv,
                                                  float* __restrict__ obuf) {
  __shared__ float lds_p[256];  // 16x16 P tile for C->A layout conversion
  const int lane = threadIdx.x;
  const int n0 = blockIdx.x * 16;  // query tile
  const int p = blockIdx.y;        // (b,t)
  const int mr = lane & 15, kh = (lane >> 4) << 1, nc = lane & 15;
  const int half8 = (lane >> 4) << 3;
  const float* base = qkv + (size_t)p * 512 * 192;

  // resident Q fragments (16 k-steps over E=64)
  v2f qf[16];
  const float* qrow = base + (size_t)(n0 + mr) * 192;
#pragma unroll
  for (int ks = 0; ks < 16; ++ks) {
    int k0 = ks * 4 + kh;
    qf[ks].x = qrow[k0];
    qf[ks].y = qrow[k0 + 1];
  }

  float m_i[8], l_i[8];
#pragma unroll
  for (int r = 0; r < 8; ++r) {
    m_i[r] = -1e30f;
    l_i[r] = 0.f;
  }
  v8f acc4[4];
#pragma unroll
  for (int et = 0; et < 4; ++et)
#pragma unroll
    for (int r = 0; r < 8; ++r) acc4[et][r] = 0.f;

  for (int j = 0; j < 32; ++j) {  // key tiles
    v8f s;
#pragma unroll
    for (int r = 0; r < 8; ++r) s[r] = 0.f;
    const float* krow = base + (size_t)(j * 16 + nc) * 192 + 64;
#pragma unroll
    for (int ks = 0; ks < 16; ++ks) {
      int k0 = ks * 4 + kh;
      v2f kf;
      kf.x = krow[k0];
      kf.y = krow[k0 + 1];
      s = wmma4(qf[ks], kf, s);
    }
    // online softmax (rows live in 16-lane halves)
#pragma unroll
    for (int r = 0; r < 8; ++r) {
      float sv = s[r] * 0.125f;  // 1/sqrt(64)
      float mx = sv;
      mx = fmaxf(mx, __shfl_xor(mx, 1));
      mx = fmaxf(mx, __shfl_xor(mx, 2));
      mx = fmaxf(mx, __shfl_xor(mx, 4));
      mx = fmaxf(mx, __shfl_xor(mx, 8));
      float nm = fmaxf(m_i[r], mx);
      float sc = __expf(m_i[r] - nm);
      float pv = __expf(sv - nm);
      float rs = pv;
      rs += __shfl_xor(rs, 1);
      rs += __shfl_xor(rs, 2);
      rs += __shfl_xor(rs, 4);
      rs += __shfl_xor(rs, 8);
      l_i[r] = l_i[r] * sc + rs;
      m_i[r] = nm;
#pragma unroll
      for (int et = 0; et < 4; ++et) acc4[et][r] *= sc;
      s[r] = pv;
    }
    __syncthreads();  // WAR on lds_p from previous tile
#pragma unroll
    for (int r = 0; r < 8; ++r) lds_p[(r + half8) * 16 + nc] = s[r];
    __syncthreads();
    v2f af[4];
#pragma unroll
    for (int ks2 = 0; ks2 < 4; ++ks2) {
      int k0 = ks2 * 4 + kh;
      af[ks2].x = lds_p[mr * 16 + k0];
      af[ks2].y = lds_p[mr * 16 + k0 + 1];
    }
#pragma unroll
    for (int ks2 = 0; ks2 < 4; ++ks2) {
      int k0 = ks2 * 4 + kh;
      const float* v1 = base + (size_t)(j * 16 + k0) * 192 + 128;
      const float* v2r = base + (size_t)(j * 16 + k0 + 1) * 192 + 128;
#pragma unroll
      for (int et = 0; et < 4; ++et) {
        v2f bf;
        bf.x = v1[et * 16 + nc];
        bf.y = v2r[et * 16 + nc];
        acc4[et] = wmma4(af[ks2], bf, acc4[et]);
      }
    }
  }
#pragma unroll
  for (int et = 0; et < 4; ++et)
#pragma unroll
    for (int r = 0; r < 8; ++r)
      obuf[((size_t)p * 512 + n0 + half8 + r) * 64 + et * 16 + nc] =
          acc4[et][r] / l_i[r];
}

// ================= 4. time conv as K'=192 WMMA GEMM =========================
// out[(b*T+t)*N+n][h] = tc_b[h] + sum_{kt,e} fc[(b,t+kt-1)][n][e]*w[h][e][kt]
// One wave computes a 16x64 tile (all H). ka = kt*64+e.
__global__ __launch_bounds__(32) void tconv_kernel(
    const float* __restrict__ fc, const float* __restrict__ tcw,
    const float* __restrict__ tcb, float* __restrict__ out) {
  const int lane = threadIdx.x;
  const int m0 = blockIdx.x * 16;
  const int bt = m0 >> 9, n0 = m0 & 511;
  const int b = bt / 24, t = bt % 24;
  const int mr = lane & 15, kh = (lane >> 4) << 1, nc = lane & 15;
  v8f acc[4];
#pragma unroll
  for (int et = 0; et < 4; ++et) {
    float bv = tcb[et * 16 + nc];
#pragma unroll
    for (int i = 0; i < 8; ++i) acc[et][i] = bv;
  }
  const float* fcr = fc + ((size_t)(b * 24) * 512 + n0 + mr) * 64;

  auto loadA = [&](int kq) {
    const int ka = kq + kh;
    const int kt1 = ka >> 6, e1 = ka & 63, tt1 = t + kt1 - 1;
    const int kt2 = (ka + 1) >> 6, e2 = (ka + 1) & 63, tt2 = t + kt2 - 1;
    v2f a;
    a.x = (tt1 >= 0 && tt1 < 24) ? fcr[(size_t)tt1 * 512 * 64 + e1] : 0.f;
    a.y = (tt2 >= 0 && tt2 < 24) ? fcr[(size_t)tt2 * 512 * 64 + e2] : 0.f;
    return a;
  };
  auto loadB = [&](int kq, v2f bb[4]) {
    const int ka = kq + kh;
    const int kt1 = ka >> 6, e1 = ka & 63;
    const int kt2 = (ka + 1) >> 6, e2 = (ka + 1) & 63;
#pragma unroll
    for (int et = 0; et < 4; ++et) {
      bb[et].x = tcw[(size_t)(et * 16 + nc) * 192 + e1 * 3 + kt1];
      bb[et].y = tcw[(size_t)(et * 16 + nc) * 192 + e2 * 3 + kt2];
    }
  };
  gemm_pipeline<192>(loadA, loadB, acc);

  const int rbase = n0 + ((lane >> 4) << 3);
#pragma unroll
  for (int et = 0; et < 4; ++et)
#pragma unroll
    for (int r = 0; r < 8; ++r)
      out[((size_t)bt * 512 + rbase + r) * 64 + et * 16 + nc] = acc[et][r];
}

// ================= 5. GRU pointwise update per timestep =====================
__global__ __launch_bounds__(256) void gru_step_kernel(
    const float* __restrict__ gi, const float* __restrict__ gh,
    float* __restrict__ hbuf, float* __restrict__ hs, int t) {
  const int idx = blockIdx.x * 256 + threadIdx.x;  // < 262144 = BN*H
  const int row = idx >> 6, h = idx & 63;
  const float* gir = gi + ((size_t)row * 24 + t) * 192;
  const float* ghr = gh + (size_t)row * 192;
  float ir = gir[h], iz = gir[64 + h], inn = gir[128 + h];
  float hr = ghr[h], hz = ghr[64 + h], hn = ghr[128 + h];
  float hp = hbuf[idx];
  float rr = 1.f / (1.f + __expf(-(ir + hr)));
  float zz = 1.f / (1.f + __expf(-(iz + hz)));
  float nn = tanhf(inn + rr * hn);
  float hnew = (1.f - zz) * nn + zz * hp;
  hbuf[idx] = hnew;
  hs[(size_t)t * 262144 + idx] = hnew;
}

__global__ __launch_bounds__(256) void zero_kernel(float* __restrict__ p,
                                                   int n) {
  int i = blockIdx.x * 256 + threadIdx.x;
  if (i < n) p[i] = 0.f;
}

// ================= 6. fuse (relu(res+tconv)) + LayerNorm ====================
__global__ __launch_bounds__(128) void fuse_ln_kernel(
    const float* __restrict__ hs, const float* __restrict__ tco,
    const float* __restrict__ g, const float* __restrict__ bta,
    float* __restrict__ out) {
  const int r = blockIdx.x * 128 + threadIdx.x;  // < 98304; r=(b*24+t)*512+n
  const int bt = r >> 9, n = r & 511;
  const int b = bt / 24, t = bt % 24;
  const float* hrow = hs + ((size_t)t * 4096 + (size_t)b * 512 + n) * 64;
  const float* trow = tco + (size_t)r * 64;
  float v[64];
  float mu = 0.f;
#pragma unroll
  for (int h = 0; h < 64; ++h) {
    v[h] = fmaxf(hrow[h] + trow[h], 0.f);
    mu += v[h];
  }
  mu *= (1.f / 64.f);
  float var = 0.f;
#pragma unroll
  for (int h = 0; h < 64; ++h) {
    float d = v[h] - mu;
    var += d * d;
  }
  var *= (1.f / 64.f);
  float rs = rsqrtf(var + 1e-5f);
#pragma unroll
  for (int h = 0; h < 64; ++h)
    out[(size_t)r * 64 + h] = (v[h] - mu) * rs * g[h] + bta[h];
}

// ============================== launcher ====================================
extern "C" void kernel_launch(void* const* d_in, const int* in_sizes, int n_in,
                              void* d_out, int out_size, void* d_ws,
                              size_t ws_size, hipStream_t stream) {
  const float* x = (const float*)d_in[0];
  const float* cheb = (const float*)d_in[1];
  const float* Theta = (const float*)d_in[2];
  const float* mlp_w1 = (const float*)d_in[3];
  const float* mlp_b1 = (const float*)d_in[4];
  const float* mlp_w2 = (const float*)d_in[5];
  const float* mlp_b2 = (const float*)d_in[6];
  // d_in[7..20]: temporal/spatial attention params -> dead code in reference
  const float* in_proj_w = (const float*)d_in[21];
  const float* in_proj_b = (const float*)d_in[22];
  const float* out_proj_w = (const float*)d_in[23];
  const float* out_proj_b = (const float*)d_in[24];
  const float* gat_fc_w = (const float*)d_in[25];
  const float* gat_fc_b = (const float*)d_in[26];
  const float* tc_w = (const float*)d_in[27];
  const float* tc_b = (const float*)d_in[28];
  const float* rc_w = (const float*)d_in[29];
  const float* rc_b = (const float*)d_in[30];
  const float* w_ih = (const float*)d_in[31];
  const float* w_hh = (const float*)d_in[32];
  const float* b_ih = (const float*)d_in[33];
  const float* b_hh = (const float*)d_in[34];
  const float* ln_g = (const float*)d_in[35];
  const float* ln_b = (const float*)d_in[36];

  float* ws = (float*)d_ws;
  // workspace layout (floats); gmat region reused for fc, qkv region for gi
  const size_t o_xpost = 0;                 // 1,572,864  (B,N,F,T)
  const size_t o_xr = o_xpost + 1572864;    // 1,572,864  (B,N,T,F)
  const size_t o_ybuf = o_xr + 1572864;     // 4,718,592  (B*K,N,F*T)
  const size_t o_gmat = o_ybuf + 4718592;   // 6,291,456  (p,N,E) / fc
  const size_t o_qkv = o_gmat + 6291456;    // 18,874,368 (p,N,192) / gi
  const size_t o_obuf = o_qkv + 18874368;   // 6,291,456
  const size_t o_obuf2 = o_obuf + 6291456;  // 6,291,456
  const size_t o_tco = o_obuf2 + 6291456;   // 6,291,456  (B,T,N,H)
  const size_t o_rbuf = o_tco + 6291456;    // 6,291,456  (B,N,T,H)
  const size_t o_gh = o_rbuf + 6291456;     // 786,432    (BN,192)
  const size_t o_h = o_gh + 786432;         // 262,144    (BN,H)
  const size_t o_hs = o_h + 262144;         // 6,291,456  (T,BN,H)

  // 1. DFT MLP -> xpost, xr
  dft_kernel<<<768, 128, 0, stream>>>(x, mlp_w1, mlp_b1, mlp_w2, mlp_b2,
                                      ws + o_xpost, ws + o_xr);
  // residual 1x1 conv: rbuf = xr @ rc_w + rc_b   (M=98304,K=16,N=64)
  gemm16x64_kernel<16, false, false><<<dim3(1, 6144), 32, 0, stream>>>(
      ws + o_xr, rc_w, rc_b, ws + o_rbuf, 16, 64, 64);
  // 2. ChebConv
  cheb_a_kernel<<<dim3(6, 32, 24), 32, 0, stream>>>(cheb, ws + o_xpost,
                                                    ws + o_ybuf);
  cheb_b_kernel<<<6144, 32, 0, stream>>>(ws + o_ybuf, Theta, ws + o_gmat);
  // 3. GAT: qkv projection (M=98304,K=64,N=192)
  gemm16x64_kernel<64, false, false><<<dim3(3, 6144), 32, 0, stream>>>(
      ws + o_gmat, in_proj_w, in_proj_b, ws + o_qkv, 64, 192, 192);
  attn_kernel<<<dim3(32, 192), 32, 0, stream>>>(ws + o_qkv, ws + o_obuf);
  gemm16x64_kernel<64, false, false><<<dim3(1, 6144), 32, 0, stream>>>(
      ws + o_obuf, out_proj_w, out_proj_b, ws + o_obuf2, 64, 64, 64);
  gemm16x64_kernel<64, false, true><<<dim3(1, 6144), 32, 0, stream>>>(
      ws + o_obuf2, gat_fc_w, gat_fc_b, ws + o_gmat /*fc*/, 64, 64, 64);
  // GRU input gates: gi = rbuf @ w_ih^T + b_ih  (reuses qkv region)
  gemm16x64_kernel<64, true, false><<<dim3(3, 6144), 32, 0, stream>>>(
      ws + o_rbuf, w_ih, b_ih, ws + o_qkv /*gi*/, 64, 64, 192);
  // 4. time conv (uses fc in gmat region)
  tconv_kernel<<<6144, 32, 0, stream>>>(ws + o_gmat, tc_w, tc_b, ws + o_tco);
  // 5. GRU scan over T
  zero_kernel<<<1024, 256, 0, stream>>>(ws + o_h, 262144);
  for (int t = 0; t < 24; ++t) {
    gemm16x64_kernel<64, true, false><<<dim3(3, 256), 32, 0, stream>>>(
        ws + o_h, w_hh, b_hh, ws + o_gh, 64, 64, 192);
    gru_step_kernel<<<1024, 256, 0, stream>>>(ws + o_qkv /*gi*/, ws + o_gh,
                                              ws + o_h, ws + o_hs, t);
  }
  // 6. fuse + LayerNorm -> d_out (B,T,N,H)
  fuse_ln_kernel<<<768, 128, 0, stream>>>(ws + o_hs, ws + o_tco, ln_g, ln_b,
                                          (float*)d_out);
}